// RelPos_38774964748732
// MI455X (gfx1250) — compile-verified
//
#include <hip/hip_runtime.h>

// RelPos pair embedding: out[i,j,:] = W[clamp((x_i - x_j) - bins[0], 0, 64)] + b
// N=1536, NBINS=65, EMBED=128. Pure streaming-store problem (~1.21 GB out,
// ~52us floor at 23.3 TB/s). WMMA (one-hot @ W) would add ~39 GFLOP of f32
// matrix work for zero byte savings, so the win is the data-movement path:
//   - CDNA5 async global->LDS copies (ASYNCcnt) stage the 33 KB W table,
//     one instruction per 512 B (INST_OFFSET advances both LDS + global addr)
//   - ds_load_b128 wave-uniform row gather from LDS
//   - non-temporal global_store_b128 so 1.2 GB write-once data doesn't rinse L2

#define N_SEQ   1536
#define NBINS   65
#define EMBED   128
#define JPB     128          // j-columns per block
#define THREADS 256          // 8 waves (wave32)

typedef float v4f __attribute__((ext_vector_type(4)));
typedef __attribute__((address_space(3))) v4f lds_v4f;

// One async copy: 32 lanes x 16 B = 512 B, at compile-time byte offset OFF
// (applied to both the LDS destination and the global source address).
#define ASYNC_CP_B128(loff, g, OFF)                                          \
    asm volatile("global_load_async_to_lds_b128 %0, %1, off offset:" #OFF    \
                 :: "v"(loff), "v"(g) : "memory")

__global__ __launch_bounds__(THREADS)
void relpos_gather_kernel(const int* __restrict__ x,
                          const float* __restrict__ bins,
                          const float* __restrict__ W,
                          const float* __restrict__ b,
                          float* __restrict__ out)
{
    // W table as v4f: 65 rows x 32 v4f = 2080 v4f = 33280 bytes of LDS
    __shared__ v4f sW[NBINS * (EMBED / 4)];
    __shared__ int sXj[JPB];

    const int tid  = threadIdx.x;
    const int lane = tid & 31;
    const int wave = tid >> 5;
    const int i    = blockIdx.y;
    const int j0   = blockIdx.x * JPB;

    // ---- stage W into LDS: 2080 v4f = 8 full 256-thread passes + 32-elem tail.
    // Single base address pair per thread; each pass is one instruction with an
    // immediate offset (stride 256 threads * 16 B = 4096 B). EXEC all-1s on every issue.
    {
        const v4f* g    = (const v4f*)W + tid;
        unsigned   loff = (unsigned)(unsigned long)(lds_v4f*)&sW[0] + (unsigned)tid * 16u;
        ASYNC_CP_B128(loff, g, 0);
        ASYNC_CP_B128(loff, g, 4096);
        ASYNC_CP_B128(loff, g, 8192);
        ASYNC_CP_B128(loff, g, 12288);
        ASYNC_CP_B128(loff, g, 16384);
        ASYNC_CP_B128(loff, g, 20480);
        ASYNC_CP_B128(loff, g, 24576);
        ASYNC_CP_B128(loff, g, 28672);
        if (tid < 32)                        // tail: exactly wave 0, full EXEC
            ASYNC_CP_B128(loff, g, 32768);
    }

    // x tile for this block's j-range (plain ds store; covered by barrier)
    if (tid < JPB) sXj[tid] = x[j0 + tid];

    const int xi   = x[i];
    const int bin0 = (int)bins[0];             // == -32 for this input
    const v4f breg = ((const v4f*)b)[lane];    // lane's 4 bias values, kept in VGPRs

    // each wave drains its own ASYNCcnt before signaling the workgroup barrier
    asm volatile("s_wait_asynccnt 0" ::: "memory");
    __syncthreads();

    // ---- stream the output: one wave = one 512 B output row, NT 128-bit stores ----
    const size_t rowbase = ((size_t)i * N_SEQ + j0) * EMBED;
#pragma unroll 4
    for (int jj = wave; jj < JPB; jj += (THREADS / 32)) {
        int d   = xi - sXj[jj];
        int idx = d - bin0;
        idx = idx < 0 ? 0 : (idx > NBINS - 1 ? NBINS - 1 : idx);   // wave-uniform

        v4f w = sW[idx * (EMBED / 4) + lane];   // ds_load_b128 row broadcast
        v4f o = w + breg;

        v4f* dst = (v4f*)(out + rowbase + (size_t)jj * EMBED) + lane;
        __builtin_nontemporal_store(o, dst);    // global_store_b128 th:NT
    }
}

extern "C" void kernel_launch(void* const* d_in, const int* in_sizes, int n_in,
                              void* d_out, int out_size, void* d_ws, size_t ws_size,
                              hipStream_t stream)
{
    const int*   x    = (const int*)  d_in[0];
    const float* bins = (const float*)d_in[1];
    const float* W    = (const float*)d_in[2];
    const float* b    = (const float*)d_in[3];
    float*       out  = (float*)d_out;

    dim3 grid(N_SEQ / JPB, N_SEQ);   // (12, 1536)
    relpos_gather_kernel<<<grid, THREADS, 0, stream>>>(x, bins, W, b, out);
}